// FixedTagLoss_20177756356941
// MI455X (gfx1250) — compile-verified
//
#include <hip/hip_runtime.h>
#include <hip/hip_bf16.h>
#include <stdint.h>

// Problem constants (match reference).
#define BB    2
#define SS    4096
#define VV    32000
#define ROWS  (BB * SS)          // 8192
#define F4ROW (VV / 4)           // 8000 float4 per row
#define STAGE_F4 64              // float4 per stage (32 lanes x 2)
#define STAGES (F4ROW / STAGE_F4) // 125 exactly
#define NBUF  4                  // buffers; pipeline depth = 3 stages (6 async loads)

typedef __attribute__((ext_vector_type(2))) float v2f;
typedef __attribute__((ext_vector_type(8))) float v8f;

// ---------------------------------------------------------------------------
// Kernel 1: per-row online log-sum-exp, streamed through LDS with CDNA5
// async global->LDS copies. One wave (32 lanes) per row; 8 waves per block.
// ---------------------------------------------------------------------------
__global__ __launch_bounds__(256)
void FixedTagLoss_lse_kernel(const float* __restrict__ logits,
                             const int*   __restrict__ labels,
                             float* __restrict__ wsA,   // w * nll per row
                             float* __restrict__ wsB) { // w per row
    __shared__ float4 tiles[8][NBUF][STAGE_F4];   // 32 KB

    const int lane = threadIdx.x & 31;
    const int wv   = threadIdx.x >> 5;
    const int row  = blockIdx.x * 8 + wv;

    const float* rowp = logits + (size_t)row * VV;
    const uint64_t gbase = (uint64_t)(uintptr_t)rowp;
    // Low 32 bits of a generic LDS pointer == LDS byte offset (flat aperture rule).
    const uint32_t lds0  = (uint32_t)(uintptr_t)&tiles[wv][0][0];
    const uint32_t laneb = (uint32_t)lane * 16u;

    // Issue one stage: 2 async b128 loads per lane (64 float4 per wave).
    auto issue = [&](int st) {
        uint32_t l0 = lds0 + (uint32_t)(st & (NBUF - 1)) * (STAGE_F4 * 16u) + laneb;
        uint64_t g0 = gbase + ((uint64_t)st * (STAGE_F4 * 16u)) + laneb;
        asm volatile("global_load_async_to_lds_b128 %0, %1, off"
                     :: "v"(l0), "v"(g0) : "memory");
        asm volatile("global_load_async_to_lds_b128 %0, %1, off"
                     :: "v"(l0 + 512u), "v"(g0 + 512u) : "memory");
    };

    // Prologue: 3 stages in flight (buffers 0..2); buffer 3 free for reuse slack.
    issue(0); issue(1); issue(2);

    float m = -3.4e38f;
    float s = 0.0f;

    for (int st = 0; st < STAGES; ++st) {
        const int ahead = (STAGES - 1) - st; // stages issued beyond st (steady: 2)
        if (ahead >= 2)      asm volatile("s_wait_asynccnt 4" ::: "memory");
        else if (ahead == 1) asm volatile("s_wait_asynccnt 2" ::: "memory");
        else                 asm volatile("s_wait_asynccnt 0" ::: "memory");

        const int buf = st & (NBUF - 1);
        float4 p = tiles[wv][buf][lane];
        float4 q = tiles[wv][buf][lane + 32];

        // Ensure this stage's LDS reads fully executed before recycling buffers.
        asm volatile("s_wait_dscnt 0" ::: "memory");
        const int nx = st + (NBUF - 1);
        if (nx < STAGES) issue(nx);

        float mx = fmaxf(fmaxf(fmaxf(p.x, p.y), fmaxf(p.z, p.w)),
                         fmaxf(fmaxf(q.x, q.y), fmaxf(q.z, q.w)));
        float mn = fmaxf(m, mx);
        float sc = __expf(m - mn);
        s = s * sc
          + __expf(p.x - mn) + __expf(p.y - mn) + __expf(p.z - mn) + __expf(p.w - mn)
          + __expf(q.x - mn) + __expf(q.y - mn) + __expf(q.z - mn) + __expf(q.w - mn);
        m = mn;
    }

    // Wave32 reduction of (m, s) pairs.
    for (int off = 16; off > 0; off >>= 1) {
        float mo = __shfl_xor(m, off, 32);
        float so = __shfl_xor(s, off, 32);
        float mn = fmaxf(m, mo);
        s = s * __expf(m - mn) + so * __expf(mo - mn);
        m = mn;
    }

    if (lane == 0) {
        const int b  = row >> 12;     // row / SS
        const int sp = row & (SS - 1);
        const int lbl = labels[row];
        const float xl = rowp[lbl];
        const float nll = (m + __logf(s)) - xl;

        // Tag coverage: position sp covered if inside any match of (11,22,33) or (44,55).
        const int* lb = labels + b * SS;
        bool cov = false;
#pragma unroll
        for (int t = sp - 2; t <= sp; ++t)
            if (t >= 0 && t + 3 <= SS &&
                lb[t] == 11 && lb[t + 1] == 22 && lb[t + 2] == 33) cov = true;
#pragma unroll
        for (int t = sp - 1; t <= sp; ++t)
            if (t >= 0 && t + 2 <= SS &&
                lb[t] == 44 && lb[t + 1] == 55) cov = true;

        const float wt = cov ? 2.0f : 1.0f;
        wsA[row] = wt * nll;
        wsB[row] = wt;
    }
}

// ---------------------------------------------------------------------------
// Kernel 2: single-wave deterministic reduction of the 8192 (w*nll, w) pairs.
// Lane partials are collapsed with V_WMMA_F32_16X16X4_F32 (ones-matmul reducer).
// ---------------------------------------------------------------------------
__global__ __launch_bounds__(32)
void FixedTagLoss_reduce_kernel(const float* __restrict__ wsA,
                                const float* __restrict__ wsB,
                                float* __restrict__ out) {
    const int l = threadIdx.x;
    const float4* a4 = (const float4*)wsA;
    const float4* b4 = (const float4*)wsB;

    float pa = 0.0f, pb = 0.0f;
#pragma unroll 8
    for (int it = 0; it < (ROWS / 4) / 32; ++it) {  // 64 iterations
        float4 va = a4[it * 32 + l];
        float4 vb = b4[it * 32 + l];
        pa += (va.x + va.y) + (va.z + va.w);
        pb += (vb.x + vb.y) + (vb.z + vb.w);
    }

    float ra, rb;
#if __has_builtin(__builtin_amdgcn_wmma_f32_16x16x4_f32)
    // A holds the 32 lane partials (bijection lane->(m,k) slot; the rest zero).
    // B = ones => D[m,n] = sum_k A[m,k]; summing D's 8 VGPRs + shfl_xor(16)
    // yields the full sum in every lane.
    v2f a; a.x = pa; a.y = 0.0f;
    v2f b; b.x = pb; b.y = 0.0f;
    v2f ones; ones.x = 1.0f; ones.y = 1.0f;
    v8f ca = {};
    v8f cb = {};
    ca = __builtin_amdgcn_wmma_f32_16x16x4_f32(false, a, false, ones,
                                               (short)0, ca, false, false);
    cb = __builtin_amdgcn_wmma_f32_16x16x4_f32(false, b, false, ones,
                                               (short)0, cb, false, false);
    ra = ((ca[0] + ca[1]) + (ca[2] + ca[3])) + ((ca[4] + ca[5]) + (ca[6] + ca[7]));
    rb = ((cb[0] + cb[1]) + (cb[2] + cb[3])) + ((cb[4] + cb[5]) + (cb[6] + cb[7]));
    ra += __shfl_xor(ra, 16, 32);
    rb += __shfl_xor(rb, 16, 32);
#else
    ra = pa; rb = pb;
    for (int off = 16; off > 0; off >>= 1) {
        ra += __shfl_xor(ra, off, 32);
        rb += __shfl_xor(rb, off, 32);
    }
#endif

    if (l == 0) out[0] = ra / rb;
}

// ---------------------------------------------------------------------------
extern "C" void kernel_launch(void* const* d_in, const int* in_sizes, int n_in,
                              void* d_out, int out_size, void* d_ws, size_t ws_size,
                              hipStream_t stream) {
    const float* logits = (const float*)d_in[0];
    const int*   labels = (const int*)d_in[1];
    float* wsA = (float*)d_ws;          // ROWS floats
    float* wsB = wsA + ROWS;            // ROWS floats
    float* out = (float*)d_out;

    FixedTagLoss_lse_kernel<<<ROWS / 8, 256, 0, stream>>>(logits, labels, wsA, wsB);
    FixedTagLoss_reduce_kernel<<<1, 32, 0, stream>>>(wsA, wsB, out);
}